// GRU_UNIT_25898652794990
// MI455X (gfx1250) — compile-verified
//
#include <hip/hip_runtime.h>
#include <math.h>

typedef float v2f __attribute__((ext_vector_type(2)));
typedef float v8f __attribute__((ext_vector_type(8)));

#define HID 4096
#define ACT_NONE 0
#define ACT_SIGMOID 1
#define ACT_TANH 2

// ---------------------------------------------------------------------------
// One 16-row x cols GEMV tile accumulated via V_WMMA_F32_16X16X4_F32.
// A tile (16x4 f32): lane(0..15)=row m, VGPRs K={k,k+1}; lane(16..31)=row m,
// K={k+2,k+3}  -> per-lane global_load_b64 of 2 consecutive weights.
// B tile (4x16 f32): broadcast x[k..k+3] across all N columns:
//   VGPR0: lanes0-15 = x[k]   (K=0), lanes16-31 = x[k+2] (K=2)
//   VGPR1: lanes0-15 = x[k+1] (K=1), lanes16-31 = x[k+3] (K=3)
// -> same {k + 2*half, +1} indexing as A; one ds_load_b64 per lane from LDS.
// Result: every N column of D holds the identical 16 dot products.
// ---------------------------------------------------------------------------
__device__ __forceinline__ v8f gemv_tile(const float* __restrict__ W,
                                         const float* __restrict__ xs,
                                         int row0, int cols, int lane, v8f c) {
  const int half = lane >> 4;          // 0: K={0,1}, 1: K={2,3}
  const int m    = lane & 15;          // row within tile
  const float* wrow = W + (size_t)(row0 + m) * (size_t)cols + 2 * half;
  const float* xp   = xs + 2 * half;
#pragma unroll 8
  for (int k = 0; k < cols; k += 4) {
    v2f a = *(const v2f*)(wrow + k);   // 8B-aligned global_load_b64
    v2f b = *(const v2f*)(xp + k);     // 8B-aligned ds_load_b64
    c = __builtin_amdgcn_wmma_f32_16x16x4_f32(
        /*neg_a=*/false, a, /*neg_b=*/false, b,
        /*c_mod=*/(short)0, c, /*reuse_a=*/false, /*reuse_b=*/false);
  }
  return c;
}

// D/C layout: lanes 0-15, VGPR v -> M=v (N=lane); lanes 16-31, VGPR v -> M=v+8.
// B was broadcast over N, so all lanes in each half hold the same rows;
// lane 0 writes rows row0+0..7, lane 16 writes rows row0+8..15.
__device__ __forceinline__ void store_tile(float* __restrict__ out,
                                           const float* __restrict__ bias,
                                           const float* __restrict__ add,
                                           int row0, int lane, int act, v8f c) {
  if ((lane & 15) == 0) {
    const int base = row0 + ((lane >> 4) << 3);
#pragma unroll
    for (int v = 0; v < 8; ++v) {
      float acc = c[v];
      if (bias) acc += bias[base + v];
      if (add)  acc += add[base + v];
      if (act == ACT_SIGMOID)      acc = 1.0f / (1.0f + __expf(-acc));
      else if (act == ACT_TANH)    acc = tanhf(acc);
      out[base + v] = acc;
    }
  }
}

// ---------------------------------------------------------------------------
// Fused gate kernel: z = sigmoid(Wxz@x + Whz@h + bz), r likewise,
// xh = Wxh@x + bh.  3 * (H/16) row-tiles in one launch for wave parallelism.
// x and h are staged once into LDS per workgroup (32 KB of 320 KB/WGP).
// ---------------------------------------------------------------------------
__global__ void __launch_bounds__(256)
gru_gates_wmma(const float* __restrict__ w_xz, const float* __restrict__ w_hz,
               const float* __restrict__ w_xr, const float* __restrict__ w_hr,
               const float* __restrict__ w_xh,
               const float* __restrict__ x, const float* __restrict__ h,
               const float* __restrict__ b_z, const float* __restrict__ b_r,
               const float* __restrict__ b_h,
               float* __restrict__ zb, float* __restrict__ rb,
               float* __restrict__ xhb, int H, int I) {
  __shared__ float xs[HID];
  __shared__ float hs[HID];
  const int tid = threadIdx.x;
  for (int i = tid; i < I; i += blockDim.x) xs[i] = x[i];
  for (int i = tid; i < H; i += blockDim.x) hs[i] = h[i];
  __syncthreads();

  const int lane    = tid & 31;
  const int wpb     = blockDim.x >> 5;
  const int gw      = blockIdx.x * wpb + (tid >> 5);
  const int gstride = gridDim.x * wpb;
  const int rowTiles = H >> 4;
  const int nTiles   = 3 * rowTiles;

  for (int t = gw; t < nTiles; t += gstride) {   // uniform per wave
    const int g    = t / rowTiles;
    const int row0 = (t - g * rowTiles) << 4;
    v8f c = {};
    if (g == 0) {
      c = gemv_tile(w_xz, xs, row0, I, lane, c);
      c = gemv_tile(w_hz, hs, row0, H, lane, c);
      store_tile(zb, b_z, nullptr, row0, lane, ACT_SIGMOID, c);
    } else if (g == 1) {
      c = gemv_tile(w_xr, xs, row0, I, lane, c);
      c = gemv_tile(w_hr, hs, row0, H, lane, c);
      store_tile(rb, b_r, nullptr, row0, lane, ACT_SIGMOID, c);
    } else {
      c = gemv_tile(w_xh, xs, row0, I, lane, c);
      store_tile(xhb, b_h, nullptr, row0, lane, ACT_NONE, c);
    }
  }
}

// Generic out = act(W@x + bias + add); used for w_hh (add=xh, tanh) and w_hy.
__global__ void __launch_bounds__(256)
gemv_wmma(const float* __restrict__ W, const float* __restrict__ x,
          const float* __restrict__ bias, const float* __restrict__ add,
          float* __restrict__ out, int rows, int cols, int act) {
  __shared__ float xs[HID];
  const int tid = threadIdx.x;
  for (int i = tid; i < cols; i += blockDim.x) xs[i] = x[i];
  __syncthreads();

  const int lane    = tid & 31;
  const int wpb     = blockDim.x >> 5;
  const int gw      = blockIdx.x * wpb + (tid >> 5);
  const int gstride = gridDim.x * wpb;
  const int nTiles  = rows >> 4;

  for (int t = gw; t < nTiles; t += gstride) {
    const int row0 = t << 4;
    v8f c = {};
    c = gemv_tile(W, xs, row0, cols, lane, c);
    store_tile(out, bias, add, row0, lane, act, c);
  }
}

__global__ void gru_rh(const float* __restrict__ r, const float* __restrict__ h,
                       float* __restrict__ rh, int n) {
  int i = blockIdx.x * blockDim.x + threadIdx.x;
  if (i < n) rh[i] = r[i] * h[i];
}

__global__ void gru_hstate(const float* __restrict__ z, const float* __restrict__ h,
                           const float* __restrict__ hhat, float* __restrict__ hout,
                           int n) {
  int i = blockIdx.x * blockDim.x + threadIdx.x;
  if (i < n) {
    float zi = z[i];
    hout[i] = (1.0f - zi) * h[i] + zi * hhat[i];
  }
}

// Single-workgroup softmax over n=32000 logits (reads ~375 KB total, trivial).
__global__ void __launch_bounds__(1024)
softmax_kernel(const float* __restrict__ logits, float* __restrict__ y, int n) {
  __shared__ float red[1024];
  const int tid = threadIdx.x;

  float m = -3.402823466e+38f;
  for (int i = tid; i < n; i += 1024) m = fmaxf(m, logits[i]);
  red[tid] = m; __syncthreads();
  for (int s = 512; s > 0; s >>= 1) {
    if (tid < s) red[tid] = fmaxf(red[tid], red[tid + s]);
    __syncthreads();
  }
  const float gmax = red[0];
  __syncthreads();

  float sum = 0.0f;
  for (int i = tid; i < n; i += 1024) sum += __expf(logits[i] - gmax);
  red[tid] = sum; __syncthreads();
  for (int s = 512; s > 0; s >>= 1) {
    if (tid < s) red[tid] += red[tid + s];
    __syncthreads();
  }
  const float inv = 1.0f / red[0];

  for (int i = tid; i < n; i += 1024) y[i] = __expf(logits[i] - gmax) * inv;
}

extern "C" void kernel_launch(void* const* d_in, const int* in_sizes, int n_in,
                              void* d_out, int out_size, void* d_ws, size_t ws_size,
                              hipStream_t stream) {
  const float* prev_h = (const float*)d_in[0];
  const float* x_t    = (const float*)d_in[1];
  const float* w_xz   = (const float*)d_in[2];
  const float* w_hz   = (const float*)d_in[3];
  const float* b_z    = (const float*)d_in[4];
  const float* w_xr   = (const float*)d_in[5];
  const float* w_hr   = (const float*)d_in[6];
  const float* b_r    = (const float*)d_in[7];
  const float* w_xh   = (const float*)d_in[8];
  const float* w_hh   = (const float*)d_in[9];
  const float* b_h    = (const float*)d_in[10];
  const float* w_hy   = (const float*)d_in[11];
  const float* b_y    = (const float*)d_in[12];

  const int H = in_sizes[0];   // 4096
  const int I = in_sizes[1];   // 4096
  const int O = in_sizes[12];  // 32000

  float* out = (float*)d_out;          // [0,H) = h_t, [H, H+O) = y_t
  float* ws  = (float*)d_ws;
  float* zb     = ws;                  // H
  float* rb     = ws + H;              // H
  float* xhb    = ws + 2 * H;          // H  (includes b_h)
  float* rhb    = ws + 3 * H;          // H
  float* hhatb  = ws + 4 * H;          // H
  float* logits = ws + 5 * H;          // O

  const int BT = 256;                  // 8 waves/block
  const int wpb = BT / 32;

  // 1) z, r, xh (fused; 3*H/16 = 768 wave-tiles)
  {
    int tiles = 3 * (H >> 4);
    int grid = (tiles + wpb - 1) / wpb;
    gru_gates_wmma<<<grid, BT, 0, stream>>>(w_xz, w_hz, w_xr, w_hr, w_xh,
                                            x_t, prev_h, b_z, b_r, b_h,
                                            zb, rb, xhb, H, I);
  }
  // 2) rh = r * prev_h
  gru_rh<<<(H + 255) / 256, 256, 0, stream>>>(rb, prev_h, rhb, H);
  // 3) h_hat = tanh(w_hh @ rh + xh)
  {
    int tiles = H >> 4;
    int grid = (tiles + wpb - 1) / wpb;
    gemv_wmma<<<grid, BT, 0, stream>>>(w_hh, rhb, nullptr, xhb, hhatb,
                                       H, H, ACT_TANH);
  }
  // 4) h_t = (1-z)*h + z*h_hat  -> d_out[0..H)
  gru_hstate<<<(H + 255) / 256, 256, 0, stream>>>(zb, prev_h, hhatb, out, H);
  // 5) logits = w_hy @ h_t + b_y   (dominant 524 MB stream; 2000 wave-tiles)
  {
    int tiles = O >> 4;
    int grid = (tiles + wpb - 1) / wpb;
    gemv_wmma<<<grid, BT, 0, stream>>>(w_hy, out, b_y, nullptr, logits,
                                       O, H, ACT_NONE);
  }
  // 6) y_t = softmax(logits) -> d_out[H..H+O)
  softmax_kernel<<<1, 1024, 0, stream>>>(logits, out + H, O);
}